// DirectedGATLayer_86758339379427
// MI455X (gfx1250) — compile-verified
//
#include <hip/hip_runtime.h>
#include <math.h>

typedef float v2f __attribute__((ext_vector_type(2)));
typedef float v4f __attribute__((ext_vector_type(4)));
typedef float v8f __attribute__((ext_vector_type(8)));

#define IN_DIM 128
#define OUT_DIM 128
#define NHEAD 4
#define HD 32
#define NEG_SLOPE 0.2f
#define EPS_SOFTMAX 1e-8f
#define EPS_LN 1e-5f

// -------------------- float atomic max via int tricks --------------------
__device__ __forceinline__ void atomicMaxF32(float* addr, float val) {
    if (val >= 0.0f) {
        atomicMax((int*)addr, __float_as_int(val));
    } else {
        atomicMin((unsigned int*)addr, (unsigned int)__float_as_int(val));
    }
}

// -------------------- init: emax = -inf, sums = 0 --------------------
__global__ void init_kernel(float* emax0, float* emax1, float* asum0, float* asum1, int NH) {
    int gid = blockIdx.x * blockDim.x + threadIdx.x;
    if (gid >= NH) return;
    emax0[gid] = -INFINITY;
    emax1[gid] = -INFINITY;
    asum0[gid] = 0.0f;
    asum1[gid] = 0.0f;
}

// -------------------- fused GEMM: h@[W | W_self] via fp32 WMMA --------------------
// Block: 256 threads = 8 waves. Block b computes rows [16b, 16b+16).
// Wave w computes combined columns [32w, 32w+32): two 16x16 WMMA tiles.
// Combined cols 0..127 -> h_proj, 128..255 -> d_out (h_self accumulator base).
__global__ __launch_bounds__(256) void gemm_wmma_kernel(
    const float* __restrict__ h, const float* __restrict__ W,
    const float* __restrict__ Wself, float* __restrict__ h_proj,
    float* __restrict__ out, int N)
{
    const int wave = threadIdx.x >> 5;
    const int lane = threadIdx.x & 31;
    const int row0 = blockIdx.x * 16;
    const int mrow = lane & 15;            // row (for A) / col (for B/C) within tile
    const int khalf = (lane >> 4) * 2;     // K sub-offset: lanes 16-31 hold K+2,K+3

    int arow_idx = row0 + mrow;
    if (arow_idx >= N) arow_idx = N - 1;   // clamp (N is a multiple of 16 here anyway)
    const float* arow = h + (size_t)arow_idx * IN_DIM;

    const int col0 = wave * 32;            // combined column base for this wave
    const int ccol0 = col0 + mrow;         // tile-0 column for this lane
    const int ccol1 = col0 + 16 + mrow;    // tile-1 column
    // wave-uniform selection: each 16-wide tile lies entirely in one matrix
    const bool t0_proj = (col0 < 128);
    const bool t1_proj = (col0 + 16 < 128);
    const float* bp0 = t0_proj ? (W + ccol0) : (Wself + (ccol0 - 128));
    const float* bp1 = t1_proj ? (W + ccol1) : (Wself + (ccol1 - 128));

    v8f c0 = {};
    v8f c1 = {};
    for (int k0 = 0; k0 < IN_DIM; k0 += 4) {
        const int k = k0 + khalf;
        v2f a, b0, b1;
        a.x = arow[k];
        a.y = arow[k + 1];
        b0.x = bp0[(size_t)k * OUT_DIM];
        b0.y = bp0[(size_t)(k + 1) * OUT_DIM];
        b1.x = bp1[(size_t)k * OUT_DIM];
        b1.y = bp1[(size_t)(k + 1) * OUT_DIM];
        c0 = __builtin_amdgcn_wmma_f32_16x16x4_f32(false, a, false, b0,
                                                   (short)0, c0, false, false);
        c1 = __builtin_amdgcn_wmma_f32_16x16x4_f32(false, a, false, b1,
                                                   (short)0, c1, false, false);
    }

    // Store: C/D layout -> lane l, VGPR v holds (M = v + (l>=16 ? 8 : 0), N = l%16)
    const int mbase = (lane >> 4) * 8;
    float* dst0 = t0_proj ? (h_proj + ccol0) : (out + (ccol0 - 128));
    float* dst1 = t1_proj ? (h_proj + ccol1) : (out + (ccol1 - 128));
    if (row0 + 16 <= N) {
        // fast path: no per-element bounds checks
#pragma unroll
        for (int v = 0; v < 8; ++v) {
            const size_t row = (size_t)(row0 + mbase + v);
            dst0[row * OUT_DIM] = c0[v];
            dst1[row * OUT_DIM] = c1[v];
        }
    } else {
#pragma unroll
        for (int v = 0; v < 8; ++v) {
            const int row = row0 + mbase + v;
            if (row < N) {
                dst0[(size_t)row * OUT_DIM] = c0[v];
                dst1[(size_t)row * OUT_DIM] = c1[v];
            }
        }
    }
}

// -------------------- per-node attention scores --------------------
// s_*[n*4+h] = dot(h_proj[n, h*32:(h+1)*32], a_vec_half)
__global__ void score_kernel(const float* __restrict__ h_proj,
                             const float* __restrict__ a_in,
                             const float* __restrict__ a_out,
                             float* __restrict__ s_in_s, float* __restrict__ s_in_r,
                             float* __restrict__ s_out_s, float* __restrict__ s_out_r,
                             int NH)
{
    int gid = blockIdx.x * blockDim.x + threadIdx.x;
    if (gid >= NH) return;
    const int hh = gid & 3;
    const int n = gid >> 2;
    const float* x = h_proj + (size_t)n * OUT_DIM + hh * HD;
    const float* ai = a_in + hh * 2 * HD;
    const float* ao = a_out + hh * 2 * HD;
    float s0 = 0.f, s1 = 0.f, s2 = 0.f, s3 = 0.f;
#pragma unroll
    for (int d = 0; d < HD; d += 4) {
        const v4f xv = *(const v4f*)(x + d);
#pragma unroll
        for (int j = 0; j < 4; ++j) {
            s0 += xv[j] * ai[d + j];
            s1 += xv[j] * ai[HD + d + j];
            s2 += xv[j] * ao[d + j];
            s3 += xv[j] * ao[HD + d + j];
        }
    }
    s_in_s[gid] = s0;
    s_in_r[gid] = s1;
    s_out_s[gid] = s2;
    s_out_r[gid] = s3;
}

// -------------------- phase A: segment max of leaky-relu logits --------------------
// one thread per edge; 4 heads via b128 score loads
__global__ void edge_max_kernel(const int* __restrict__ snd, const int* __restrict__ rcv,
                                const float* __restrict__ s_s, const float* __restrict__ s_r,
                                float* __restrict__ emax, int E)
{
    int e = blockIdx.x * blockDim.x + threadIdx.x;
    if (e >= E) return;
    const int s = snd[e];
    const int r = rcv[e];
    const v4f ss = *(const v4f*)(s_s + (size_t)s * 4);
    const v4f sr = *(const v4f*)(s_r + (size_t)r * 4);
#pragma unroll
    for (int hh = 0; hh < NHEAD; ++hh) {
        float ev = ss[hh] + sr[hh];
        ev = (ev > 0.0f) ? ev : NEG_SLOPE * ev;
        atomicMaxF32(&emax[(size_t)r * 4 + hh], ev);
    }
}

// -------------------- phase B: alpha = exp(e - max), segment sum --------------------
__global__ void edge_sum_kernel(const int* __restrict__ snd, const int* __restrict__ rcv,
                                const float* __restrict__ s_s, const float* __restrict__ s_r,
                                const float* __restrict__ emax,
                                float* __restrict__ alpha_buf, float* __restrict__ asum,
                                int E)
{
    int e = blockIdx.x * blockDim.x + threadIdx.x;
    if (e >= E) return;
    const int s = snd[e];
    const int r = rcv[e];
    const v4f ss = *(const v4f*)(s_s + (size_t)s * 4);
    const v4f sr = *(const v4f*)(s_r + (size_t)r * 4);
    const v4f em = *(const v4f*)(emax + (size_t)r * 4);
    v4f alpha;
#pragma unroll
    for (int hh = 0; hh < NHEAD; ++hh) {
        float ev = ss[hh] + sr[hh];
        ev = (ev > 0.0f) ? ev : NEG_SLOPE * ev;
        alpha[hh] = __expf(ev - em[hh]);
        atomicAdd(&asum[(size_t)r * 4 + hh], alpha[hh]);
    }
    *(v4f*)(alpha_buf + (size_t)e * 4) = alpha;   // b128 store
}

// -------------------- phase C: weighted scatter-add aggregation --------------------
// one thread per (edge, head): 8x b128 gathers + 32 atomic f32 adds
__global__ void edge_agg_kernel(const int* __restrict__ snd, const int* __restrict__ rcv,
                                const float* __restrict__ h_proj,
                                const float* __restrict__ alpha_buf,
                                const float* __restrict__ asum,
                                float* __restrict__ out, int EH)
{
    int gid = blockIdx.x * blockDim.x + threadIdx.x;
    if (gid >= EH) return;
    const int hh = gid & 3;
    const int e = gid >> 2;
    const int s = snd[e];
    const int r = rcv[e];
    const float w = alpha_buf[(size_t)e * 4 + hh] / (asum[(size_t)r * 4 + hh] + EPS_SOFTMAX);
    const float* xs = h_proj + (size_t)s * OUT_DIM + hh * HD;
    float* xo = out + (size_t)r * OUT_DIM + hh * HD;
#pragma unroll
    for (int d = 0; d < HD; d += 4) {
        const v4f xv = *(const v4f*)(xs + d);
#pragma unroll
        for (int j = 0; j < 4; ++j) {
            atomicAdd(xo + d + j, w * xv[j]);
        }
    }
}

// -------------------- LayerNorm over combined + bias --------------------
__global__ __launch_bounds__(128) void ln_kernel(float* __restrict__ out,
                                                 const float* __restrict__ bias,
                                                 const float* __restrict__ gamma,
                                                 const float* __restrict__ beta,
                                                 int N)
{
    __shared__ float red[OUT_DIM];
    const int n = blockIdx.x;
    const int t = threadIdx.x;
    if (n >= N) return;
    const float x = out[(size_t)n * OUT_DIM + t] + bias[t];
    red[t] = x;
    __syncthreads();
    for (int s = OUT_DIM / 2; s > 0; s >>= 1) {
        if (t < s) red[t] += red[t + s];
        __syncthreads();
    }
    const float mean = red[0] * (1.0f / OUT_DIM);
    __syncthreads();
    const float d = x - mean;
    red[t] = d * d;
    __syncthreads();
    for (int s = OUT_DIM / 2; s > 0; s >>= 1) {
        if (t < s) red[t] += red[t + s];
        __syncthreads();
    }
    const float var = red[0] * (1.0f / OUT_DIM);
    const float inv = rsqrtf(var + EPS_LN);
    out[(size_t)n * OUT_DIM + t] = d * inv * gamma[t] + beta[t];
}

// -------------------- launch --------------------
extern "C" void kernel_launch(void* const* d_in, const int* in_sizes, int n_in,
                              void* d_out, int out_size, void* d_ws, size_t ws_size,
                              hipStream_t stream) {
    const float* h      = (const float*)d_in[0];
    const int*   eidx   = (const int*)d_in[1];
    const float* W      = (const float*)d_in[2];
    const float* Wself  = (const float*)d_in[3];
    const float* a_in   = (const float*)d_in[4];
    const float* a_out  = (const float*)d_in[5];
    const float* bias   = (const float*)d_in[6];
    const float* gamma  = (const float*)d_in[7];
    const float* beta   = (const float*)d_in[8];
    float* out = (float*)d_out;

    const int N = in_sizes[0] / IN_DIM;
    const int E = in_sizes[1] / 2;
    const int NH = N * NHEAD;
    const int EH = E * NHEAD;

    const int* src = eidx;        // edge_index[0]
    const int* dst = eidx + E;    // edge_index[1]

    // workspace layout (floats); all segments 16B-aligned
    float* ws = (float*)d_ws;
    float* h_proj  = ws;                                  // N*128
    float* s_in_s  = h_proj + (size_t)N * OUT_DIM;        // N*4 each
    float* s_in_r  = s_in_s + NH;
    float* s_out_s = s_in_r + NH;
    float* s_out_r = s_out_s + NH;
    float* emax0   = s_out_r + NH;
    float* emax1   = emax0 + NH;
    float* asum0   = emax1 + NH;
    float* asum1   = asum0 + NH;
    float* alpha0  = asum1 + NH;                          // E*4 each
    float* alpha1  = alpha0 + (size_t)E * NHEAD;

    // 1) init maxima / sums
    {
        int threads = 256, blocks = (NH + threads - 1) / threads;
        init_kernel<<<blocks, threads, 0, stream>>>(emax0, emax1, asum0, asum1, NH);
    }
    // 2) WMMA GEMM: h_proj and h_self (h_self seeds d_out accumulator)
    {
        int blocks = (N + 15) / 16;
        gemm_wmma_kernel<<<blocks, 256, 0, stream>>>(h, W, Wself, h_proj, out, N);
    }
    // 3) per-node attention scores
    {
        int threads = 256, blocks = (NH + threads - 1) / threads;
        score_kernel<<<blocks, threads, 0, stream>>>(h_proj, a_in, a_out,
                                                     s_in_s, s_in_r, s_out_s, s_out_r, NH);
    }
    // 4) segment max, both directions (one thread per edge)
    {
        int threads = 256, blocks = (E + threads - 1) / threads;
        // in-direction: sender=src, receiver=dst
        edge_max_kernel<<<blocks, threads, 0, stream>>>(src, dst, s_in_s, s_in_r, emax0, E);
        // out-direction: sender=dst, receiver=src
        edge_max_kernel<<<blocks, threads, 0, stream>>>(dst, src, s_out_s, s_out_r, emax1, E);
    }
    // 5) exp + segment sum (caches unnormalized alpha)
    {
        int threads = 256, blocks = (E + threads - 1) / threads;
        edge_sum_kernel<<<blocks, threads, 0, stream>>>(src, dst, s_in_s, s_in_r, emax0,
                                                        alpha0, asum0, E);
        edge_sum_kernel<<<blocks, threads, 0, stream>>>(dst, src, s_out_s, s_out_r, emax1,
                                                        alpha1, asum1, E);
    }
    // 6) weighted scatter-add aggregation into d_out (one thread per edge-head)
    {
        int threads = 256, blocks = (EH + threads - 1) / threads;
        edge_agg_kernel<<<blocks, threads, 0, stream>>>(src, dst, h_proj,
                                                        alpha0, asum0, out, EH);
        edge_agg_kernel<<<blocks, threads, 0, stream>>>(dst, src, h_proj,
                                                        alpha1, asum1, out, EH);
    }
    // 7) bias + LayerNorm (in place on d_out)
    {
        ln_kernel<<<N, OUT_DIM, 0, stream>>>(out, bias, gamma, beta, N);
    }
}